// DSAC_15822659519167
// MI455X (gfx1250) — compile-verified
//
#include <hip/hip_runtime.h>

// DSAC plane-consensus kernel for MI455X (gfx1250, wave32).
// Hot loop: V_WMMA_F32_16X16X4_F32 computes 16 planes x 16 points signed
// distances (K=4: k*[nx,ny,nz,d] x [x,y,z,1]). The sqrt(50/ln2) factor is
// pre-folded into the plane coefficients so the per-element epilogue is just
// v_pk_mul (with neg source modifier) + v_exp_f32 + v_pk_add.
// Trans-pipe (v_exp_f32) bound by design; VALU co-executes.

typedef float v2f __attribute__((ext_vector_type(2)));
typedef float v8f __attribute__((ext_vector_type(8)));

#define ALPHA 0.5f
#define LOG2E 1.4426950408889634f
// sqrt(50 * log2(e)) = sqrt(72.13475204444817)
#define DIST_PRESCALE 8.493218671f

__device__ __forceinline__ float exp_e(float x) {
    return __builtin_amdgcn_exp2f(x * LOG2E);   // v_exp_f32 is base-2
}

// ---------------------------------------------------------------------------
// Kernel 1: fit planes from sampled triangles; compute per-plane loss vs target
// planes[b,h] = normalize([cross(p1-p0,p2-p0), -n.p0])  (all-zero -> ones)
// ---------------------------------------------------------------------------
__global__ __launch_bounds__(256) void fit_planes_kernel(
    const float* __restrict__ pts,        // (B,N,3)
    const float* __restrict__ target,     // (B,3)
    const int*   __restrict__ sample_idx, // (B,H,3)
    float* __restrict__ planes,           // (B*H,4)
    float* __restrict__ loss,             // (B*H)
    int B, int H, int N)
{
    int tid = blockIdx.x * blockDim.x + threadIdx.x;
    if (tid >= B * H) return;
    int b = tid / H;

    const int* si = sample_idx + (size_t)tid * 3;
    const float* pb = pts + (size_t)b * N * 3;
    const float* q0 = pb + (size_t)si[0] * 3;
    const float* q1 = pb + (size_t)si[1] * 3;
    const float* q2 = pb + (size_t)si[2] * 3;

    float p0x = q0[0], p0y = q0[1], p0z = q0[2];
    float e1x = q1[0] - p0x, e1y = q1[1] - p0y, e1z = q1[2] - p0z;
    float e2x = q2[0] - p0x, e2y = q2[1] - p0y, e2z = q2[2] - p0z;

    float nx = e1y * e2z - e1z * e2y;
    float ny = e1z * e2x - e1x * e2z;
    float nz = e1x * e2y - e1y * e2x;
    float d  = -(nx * p0x + ny * p0y + nz * p0z);

    if (nx == 0.f && ny == 0.f && nz == 0.f && d == 0.f) {
        nx = 1.f; ny = 1.f; nz = 1.f; d = 1.f;   // reference all_zero -> ones
    }
    float inv = rsqrtf(nx * nx + ny * ny + nz * nz);
    nx *= inv; ny *= inv; nz *= inv; d *= inv;

    float4 pv = make_float4(nx, ny, nz, d);
    *(float4*)(planes + (size_t)tid * 4) = pv;

    float tx = target[b * 3 + 0], ty = target[b * 3 + 1], tz = target[b * 3 + 2];
    float lm = (nx - tx) * (nx - tx) + (ny - ty) * (ny - ty) + (nz - tz) * (nz - tz);
    float lp = (nx + tx) * (nx + tx) + (ny + ty) * (ny + ty) + (nz + tz) * (nz + tz);
    loss[tid] = fminf(lm, lp);
}

// ---------------------------------------------------------------------------
// Kernel 2: score[b,h] = sum_n exp(-(n.p+d)^2/(2 sigma2)) via WMMA 16x16x4 f32
// Block = 8 waves, handles one (b, 16-plane tile). Wave w owns N/8 points.
// A operand (16x4): lanes 0-15 -> k*{nx,ny}, lanes 16-31 -> k*{nz,d}
// B operand (4x16): lanes 0-15 -> {x,y},  lanes 16-31 -> {z,1}
// D (16x16 f32, 8 VGPRs): v_r = row r (lanes 0-15) / row r+8 (lanes 16-31)
// With k = sqrt(50/ln2), score elem = exp2(-(D*D)).
// ---------------------------------------------------------------------------
__global__ __launch_bounds__(256) void score_kernel(
    const float* __restrict__ pts,     // (B,N,3)
    const float* __restrict__ planes,  // (B*H,4)
    float* __restrict__ score,         // (B*H)
    int H, int N, int tiles_per_wave)
{
    const int mb   = blockIdx.x;            // plane tile within batch
    const int b    = blockIdx.y;
    const int lane = threadIdx.x & 31;
    const int wave = threadIdx.x >> 5;
    const int col  = lane & 15;
    const bool hi  = lane >= 16;

    // A: 16 planes x 4 coefficients, pre-scaled by sqrt(50/ln2)
    float4 pv = *(const float4*)(planes + ((size_t)b * H + mb * 16 + col) * 4);
    v2f A;
    A[0] = (hi ? pv.z : pv.x) * DIST_PRESCALE;
    A[1] = (hi ? pv.w : pv.y) * DIST_PRESCALE;

    const float* pb = pts + (size_t)b * N * 3;
    const int nbase = wave * tiles_per_wave * 16;

    v8f acc = {};
    for (int t = 0; t < tiles_per_wave; ++t) {
        int n = nbase + t * 16 + col;
        const float* p = pb + (size_t)n * 3;
        float x = p[0], y = p[1], z = p[2];
        v2f Bm;
        Bm[0] = hi ? z : x;
        Bm[1] = hi ? 1.0f : y;
        v8f c = {};
        v8f dist = __builtin_amdgcn_wmma_f32_16x16x4_f32(
            /*neg_a=*/false, A, /*neg_b=*/false, Bm,
            /*c_mod=*/(short)0, c, /*reuse_a=*/false, /*reuse_b=*/false);
#pragma unroll
        for (int r = 0; r < 8; ++r) {
            // exp(-50*d^2) == exp2(-(k*d)^2); neg folds into the mul modifier
            acc[r] += __builtin_amdgcn_exp2f(-(dist[r] * dist[r]));
        }
    }

    // Reduce across the 16 lanes of each half (they hold distinct N columns)
#pragma unroll
    for (int off = 1; off < 16; off <<= 1) {
#pragma unroll
        for (int r = 0; r < 8; ++r)
            acc[r] += __shfl_xor(acc[r], off, 32);
    }

    // Cross-wave reduce in LDS; one block owns this (b, mb) tile exclusively
    __shared__ float ssum[16];
    if (threadIdx.x < 16) ssum[threadIdx.x] = 0.f;
    __syncthreads();
    if (col == 0) {                         // lanes 0 and 16 of every wave
#pragma unroll
        for (int r = 0; r < 8; ++r) {
            int mrow = hi ? (r + 8) : r;
            atomicAdd(&ssum[mrow], acc[r]); // ds_add_f32
        }
    }
    __syncthreads();
    if (threadIdx.x < 16)
        score[(size_t)b * H + mb * 16 + threadIdx.x] = ssum[threadIdx.x];
}

// ---------------------------------------------------------------------------
// Kernel 3: per-batch argmax (first-index tie-break), stable softmax, outputs
// out[0..B) = exp_loss ; out[B..2B) = top_loss ; out[2B..5B) = pred (B,3)
// ---------------------------------------------------------------------------
__global__ __launch_bounds__(256) void finalize_kernel(
    const float* __restrict__ planes,
    const float* __restrict__ loss,
    const float* __restrict__ score,
    float* __restrict__ out, int B, int H)
{
    int b = blockIdx.x;
    int h = threadIdx.x;

    __shared__ float sv[256];
    __shared__ int   si[256];
    __shared__ float sw[256];
    __shared__ float swl[256];

    float s = score[(size_t)b * H + h];
    sv[h] = s; si[h] = h;
    __syncthreads();
    for (int off = 128; off > 0; off >>= 1) {
        if (h < off) {
            float ov = sv[h + off]; int oi = si[h + off];
            if (ov > sv[h] || (ov == sv[h] && oi < si[h])) { sv[h] = ov; si[h] = oi; }
        }
        __syncthreads();
    }
    float smax = sv[0];
    int   midx = si[0];

    float l = loss[(size_t)b * H + h];
    float w = exp_e(ALPHA * (s - smax));     // stable softmax
    sw[h] = w; swl[h] = w * l;
    __syncthreads();
    for (int off = 128; off > 0; off >>= 1) {
        if (h < off) { sw[h] += sw[h + off]; swl[h] += swl[h + off]; }
        __syncthreads();
    }

    if (h == 0) {
        out[b]     = swl[0] / sw[0];                     // exp_loss
        out[B + b] = loss[(size_t)b * H + midx];         // top_loss
        const float* pn = planes + ((size_t)b * H + midx) * 4;
        out[2 * B + b * 3 + 0] = pn[0];                  // pred
        out[2 * B + b * 3 + 1] = pn[1];
        out[2 * B + b * 3 + 2] = pn[2];
    }
}

extern "C" void kernel_launch(void* const* d_in, const int* in_sizes, int n_in,
                              void* d_out, int out_size, void* d_ws, size_t ws_size,
                              hipStream_t stream) {
    const float* pts        = (const float*)d_in[0];  // (B,N,3) f32
    const float* target     = (const float*)d_in[1];  // (B,3)   f32
    const int*   sample_idx = (const int*)d_in[2];    // (B,H,3) i32
    float* out = (float*)d_out;

    const int B  = in_sizes[1] / 3;            // 32
    const int H  = (in_sizes[2] / 3) / B;      // 256
    const int N  = in_sizes[0] / (3 * B);      // 8192

    float* planes = (float*)d_ws;               // B*H*4 floats
    float* loss   = planes + (size_t)B * H * 4; // B*H floats
    float* score  = loss + (size_t)B * H;       // B*H floats

    fit_planes_kernel<<<(B * H + 255) / 256, 256, 0, stream>>>(
        pts, target, sample_idx, planes, loss, B, H, N);

    const int NWAVES = 8;
    const int tiles_per_wave = N / (NWAVES * 16);   // 64
    dim3 grid(H / 16, B);                           // (16, 32) blocks
    score_kernel<<<grid, NWAVES * 32, 0, stream>>>(
        pts, planes, score, H, N, tiles_per_wave);

    finalize_kernel<<<B, H, 0, stream>>>(planes, loss, score, out, B, H);
}